// RungeKuttaIntegratorCell_4277787427396
// MI455X (gfx1250) — compile-verified
//
#include <hip/hip_runtime.h>
#include <hip/hip_bf16.h>

typedef __attribute__((ext_vector_type(2))) float v2f;
typedef __attribute__((ext_vector_type(8))) float v8f;

#define DT_C   0.1f
#define CPFAS  1e-4f

// ---------------------------------------------------------------------------
// 8x8 * 8x8 f32 matmul via V_WMMA_F32_16X16X4_F32 (zero-padded to 16x16, K=8
// split into two K=4 WMMAs). L: left matrix (row-major 8x8, any aspace),
// Rm: right matrix (row-major 8x8), outm: result (row-major 8x8).
// Operand layouts (ISA 7.12.2):
//   A 16x4 f32 : lanes 0-15 row M=lane  {v0=K0, v1=K1}; lanes 16-31 {v0=K2, v1=K3}
//   B 4x16 f32 : lanes 0-15 col N=lane  {v0=K0, v1=K1}; lanes 16-31 {v0=K2, v1=K3}
//   D 16x16 f32: vgpr g, lanes 0-15 -> (M=g, N=lane); lanes 16-31 -> (M=8+g)
// Loads are made unconditional (index clamped to the valid 8x8 range) and the
// zero padding is applied with value selects -> no EXEC divergence before the
// WMMAs (straight-line ds_load/global_load + v_cndmask).
// ---------------------------------------------------------------------------
__device__ __forceinline__ void mm8_wmma(const float* L, const float* Rm, float* outm) {
    const int lane = threadIdx.x & 31;
    const int half = lane >> 4;     // 0: K pair {0,1}, 1: K pair {2,3}
    const int lr   = lane & 15;     // A: row, B/D: column
    const int lc   = lr & 7;        // clamped row/col (always in-bounds)
    const int ak   = 2 * half;
    const float m  = (lr < 8) ? 1.0f : 0.0f;   // zero-pad mask (v_cndmask)

    v2f alo, ahi, blo, bhi;
    alo.x = m * L[lc * 8 + ak + 0];
    alo.y = m * L[lc * 8 + ak + 1];
    ahi.x = m * L[lc * 8 + 4 + ak + 0];
    ahi.y = m * L[lc * 8 + 4 + ak + 1];

    blo.x = m * Rm[(ak + 0) * 8 + lc];
    blo.y = m * Rm[(ak + 1) * 8 + lc];
    bhi.x = m * Rm[(4 + ak + 0) * 8 + lc];
    bhi.y = m * Rm[(4 + ak + 1) * 8 + lc];

    v8f acc = {};
    acc = __builtin_amdgcn_wmma_f32_16x16x4_f32(false, alo, false, blo,
                                                (short)0, acc, false, false);
    acc = __builtin_amdgcn_wmma_f32_16x16x4_f32(false, ahi, false, bhi,
                                                (short)0, acc, false, false);

    // Rows 0..7 live in vgprs 0..7 of the lower half-wave (half==0).
    if (half == 0 && lr < 8) {
#pragma unroll
        for (int g = 0; g < 8; ++g) outm[g * 8 + lr] = acc[g];
    }
}

// ---------------------------------------------------------------------------
// Kernel A: per-timestep scalars + per-step propagator R_t = P4(a_t * M).
// RK4 on the linear system y' = c*M*y is exactly y_next = P4(a*M)*y with
// a = DT*c_eaq and P4 the degree-4 Taylor polynomial of exp.
// ---------------------------------------------------------------------------
__global__ void k_precompute(const float* __restrict__ log_k,
                             const float* __restrict__ u,
                             float* __restrict__ HK,     // T x 8
                             float* __restrict__ Rall,   // T x 64
                             int T) {
    int t = blockIdx.x * blockDim.x + threadIdx.x;
    if (t >= T) return;

    float k[7];
#pragma unroll
    for (int i = 0; i < 7; ++i) k[i] = exp10f(log_k[i]);

    float c_cl  = u[2 * t + 0];
    float c_so3 = u[2 * t + 1];

    // gen_eaq (c_oh_m = 10^(-14+7) = 1e-7)
    const float c_oh_m = 1e-7f;
    const float e185   = 0.032f * 55.5f;
    float sig185 = e185 + 3200.0f * c_oh_m + 3500.0f * c_cl + 200.0f * c_so3
                 + 1000.0f * CPFAS;
    float sig254 = 0.0001f * 55.5f + 20.0f * c_so3 + 10.0f * CPFAS;
    float f_h2o = e185 / sig185;
    float f_ohm = 3200.0f * c_oh_m / sig185;
    float f_cl  = 3500.0f * c_cl / sig185;
    float f_so3 = 200.0f * c_so3 / sig185;
    float f_so3_254 = 20.0f * c_so3 / sig254;
    float t_h2o = f_h2o * 0.3f * (1.0f - exp10f(-e185));
    float t_ohm = f_ohm * 0.1f * (1.0f - exp10f(-3200.0f * c_oh_m));
    float t_cl  = f_cl  * 0.4f * (1.0f - exp10f(-3500.0f * c_cl));
    float t_so3 = f_so3 * 0.4f * (1.0f - exp10f(-200.0f * c_so3));
    float num = 1e-6f * (t_h2o + t_ohm + t_cl + t_so3)
              + 1e-5f * f_so3_254 * 0.3f * (1.0f - exp10f(-20.0f * c_so3));
    float den = k[0] * CPFAS + 25700.0f + 1.5e6f * c_so3 + 1.0e6f * c_cl;
    float a = DT_C * (num / den);   // a = DT * c_eaq

#pragma unroll
    for (int i = 0; i < 7; ++i) HK[t * 8 + i] = a * k[i];
    HK[t * 8 + 7] = 0.0f;

    // R = I + aM(I + a/2 M(I + a/3 M(I + a/4 M)))  (Horner; M sparse)
    float X[64];
#pragma unroll
    for (int i = 0; i < 64; ++i) X[i] = (i % 9 == 0) ? 1.0f : 0.0f;
    const float coef[4] = {0.25f, 1.0f / 3.0f, 0.5f, 1.0f};
#pragma unroll
    for (int sj = 0; sj < 4; ++sj) {
        float s = a * coef[sj];
        float Yn[64];
#pragma unroll
        for (int col = 0; col < 8; ++col) {
            float x[7];
#pragma unroll
            for (int i = 0; i < 7; ++i) x[i] = X[i * 8 + col];
            float mx[8];
            mx[0] = -k[0] * x[0];
#pragma unroll
            for (int i = 1; i < 7; ++i) mx[i] = k[i - 1] * x[i - 1] - k[i] * x[i];
            float sum = 0.0f;
#pragma unroll
            for (int j = 0; j < 7; ++j) sum += k[j] * x[j];
            mx[7] = 2.0f * sum;
#pragma unroll
            for (int i = 0; i < 8; ++i)
                Yn[i * 8 + col] = ((i == col) ? 1.0f : 0.0f) + s * mx[i];
        }
#pragma unroll
        for (int i = 0; i < 64; ++i) X[i] = Yn[i];
    }
#pragma unroll
    for (int i = 0; i < 64; ++i) Rall[t * 64 + i] = X[i];
}

// ---------------------------------------------------------------------------
// Kernel B: chunk products G_c = R_{cL+L-1} * ... * R_{cL}  (one wave / chunk)
// ---------------------------------------------------------------------------
__global__ void k_chunkprod(const float* __restrict__ Rall,
                            float* __restrict__ G, int L) {
    __shared__ float bufA[64], bufB[64];
    int c = blockIdx.x;
    int lane = threadIdx.x;
    const float* Rc = Rall + (size_t)c * L * 64;

    bufA[lane]      = Rc[lane];
    bufA[lane + 32] = Rc[lane + 32];
    __syncthreads();

    float* cur = bufA;
    float* nxt = bufB;
    for (int j = 1; j < L; ++j) {
        mm8_wmma(Rc + j * 64, cur, nxt);
        __syncthreads();
        float* tp = cur; cur = nxt; nxt = tp;
    }
    G[c * 64 + lane]      = cur[lane];
    G[c * 64 + lane + 32] = cur[lane + 32];
}

// ---------------------------------------------------------------------------
// Kernel C: prefix propagators P_0 = I, P_c = G_{c-1} * P_{c-1}  (one wave)
// ---------------------------------------------------------------------------
__global__ void k_prefix(const float* __restrict__ G,
                         float* __restrict__ P, int C) {
    __shared__ float bufA[64], bufB[64];
    int lane = threadIdx.x;
#pragma unroll
    for (int e = lane; e < 64; e += 32) {
        float v = ((e >> 3) == (e & 7)) ? 1.0f : 0.0f;
        bufA[e] = v;
        P[e]    = v;
    }
    __syncthreads();
    float* cur = bufA;
    float* nxt = bufB;
    for (int c = 1; c < C; ++c) {
        mm8_wmma(G + (size_t)(c - 1) * 64, cur, nxt);
        __syncthreads();
        float* tp = cur; cur = nxt; nxt = tp;
        P[(size_t)c * 64 + lane]      = cur[lane];
        P[(size_t)c * 64 + lane + 32] = cur[lane + 32];
    }
}

// ---------------------------------------------------------------------------
// Kernel D: main RK4 sweep. Block = 256 batch cols, grid.y = chunk index.
// ---------------------------------------------------------------------------
__device__ __forceinline__ void fun_g(const float* g, const float* y, float* d) {
    float r[7];
#pragma unroll
    for (int i = 0; i < 7; ++i) r[i] = g[i] * y[i];
    d[0] = -r[0];
#pragma unroll
    for (int i = 1; i < 7; ++i) d[i] = r[i - 1] - r[i];
    float s = ((r[0] + r[1]) + (r[2] + r[3])) + ((r[4] + r[5]) + r[6]);
    d[7] = 2.0f * s;
}

__global__ void k_main(const float* __restrict__ y0,
                       const float* __restrict__ HK,
                       const float* __restrict__ P,
                       float* __restrict__ out, int B, int L) {
    int b = blockIdx.x * blockDim.x + threadIdx.x;
    int c = blockIdx.y;

    const float* Pc = P + (size_t)c * 64;   // block-uniform -> scalar loads

    float yi[8];
#pragma unroll
    for (int i = 0; i < 8; ++i) yi[i] = y0[(size_t)b * 8 + i];

    float y[8];
#pragma unroll
    for (int i = 0; i < 8; ++i) {
        float acc = 0.0f;
#pragma unroll
        for (int j = 0; j < 8; ++j) acc = fmaf(Pc[i * 8 + j], yi[j], acc);
        y[i] = acc;
    }

    int t0 = c * L;
    for (int j = 0; j < L; ++j) {
        int t = t0 + j;
        const float* hk = HK + (size_t)t * 8;  // block-uniform -> scalar loads
        float g[7];
#pragma unroll
        for (int i = 0; i < 7; ++i) g[i] = hk[i];

        float s1[8], s2[8], s3[8], s4[8], yt[8];
        fun_g(g, y, s1);
#pragma unroll
        for (int i = 0; i < 8; ++i) yt[i] = fmaf(0.5f, s1[i], y[i]);
        fun_g(g, yt, s2);
#pragma unroll
        for (int i = 0; i < 8; ++i) yt[i] = fmaf(0.5f, s2[i], y[i]);
        fun_g(g, yt, s3);
#pragma unroll
        for (int i = 0; i < 8; ++i) yt[i] = y[i] + s3[i];
        fun_g(g, yt, s4);
#pragma unroll
        for (int i = 0; i < 8; ++i)
            y[i] += (s1[i] + s4[i] + 2.0f * (s2[i] + s3[i])) * (1.0f / 6.0f);

        size_t base = ((size_t)t * B + b) * 5;
        out[base + 0] = y[0];
        out[base + 1] = y[2];
        out[base + 2] = y[4];
        out[base + 3] = y[5];
        out[base + 4] = y[6];
    }
}

// ---------------------------------------------------------------------------
extern "C" void kernel_launch(void* const* d_in, const int* in_sizes, int n_in,
                              void* d_out, int out_size, void* d_ws, size_t ws_size,
                              hipStream_t stream) {
    const float* log_k = (const float*)d_in[0];
    const float* u     = (const float*)d_in[1];
    const float* y0    = (const float*)d_in[2];
    float*       out   = (float*)d_out;

    int T = in_sizes[1] / 2;   // 1024
    int B = in_sizes[2] / 8;   // 4096
    const int L = 16;
    int C = T / L;             // 64

    float* ws   = (float*)d_ws;
    float* HK   = ws;                        // T*8   floats
    float* Rall = HK + (size_t)T * 8;        // T*64  floats
    float* G    = Rall + (size_t)T * 64;     // C*64  floats
    float* P    = G + (size_t)C * 64;        // C*64  floats

    k_precompute<<<dim3((T + 63) / 64), dim3(64), 0, stream>>>(log_k, u, HK, Rall, T);
    k_chunkprod<<<dim3(C), dim3(32), 0, stream>>>(Rall, G, L);
    k_prefix<<<dim3(1), dim3(32), 0, stream>>>(G, P, C);
    k_main<<<dim3(B / 256, C), dim3(256), 0, stream>>>(y0, HK, P, out, B, L);
}